// DepatchSampling_43654047597204
// MI455X (gfx1250) — compile-verified
//
#include <hip/hip_runtime.h>

typedef __attribute__((ext_vector_type(2))) float v2f;
typedef __attribute__((ext_vector_type(8))) float v8f;
typedef __attribute__((ext_vector_type(4))) int   v4i_;

#define B_      32
#define C_      64
#define L_      4096
#define PS_     16
#define STR_    8
#define PC_     511
#define HID_    64
#define ROWPAD  4112   // 4096 + 16 pad floats (covers padding patch 511)

#if defined(__gfx1250__) && __has_builtin(__builtin_amdgcn_global_load_async_to_lds_b128) && __has_builtin(__builtin_amdgcn_s_wait_asynccnt)
#define ASYNC_STAGE 1
#else
#define ASYNC_STAGE 0
#endif

#define AS_GLOBAL __attribute__((address_space(1)))
#define AS_LDS    __attribute__((address_space(3)))

// Exact gelu: 0.5*x*(1+erf(x/sqrt(2))) with branch-free A&S 7.1.26 erf
// (|abs err| < 1.5e-7). ~12 straight-line VALU ops incl. one rcp + one exp,
// no EXEC divergence (keeps the WMMA region clean).
__device__ __forceinline__ float gelu_exact(float x) {
    float z  = x * 0.70710678118654752f;
    float az = fabsf(z);
    float t  = __builtin_amdgcn_rcpf(fmaf(0.3275911f, az, 1.0f));
    float p  = 1.061405429f;
    p = fmaf(p, t, -1.453152027f);
    p = fmaf(p, t,  1.421413741f);
    p = fmaf(p, t, -0.284496736f);
    p = fmaf(p, t,  0.254829592f);
    p = p * t;
    float e    = __expf(-az * az);
    float erfa = fmaf(-p, e, 1.0f);       // erf(|z|)
    float erfz = copysignf(erfa, z);
    return 0.5f * x * (1.0f + erfz);
}

__global__ __launch_bounds__(256) void depatch_sampling_kernel(
    const float* __restrict__ X,  const float* __restrict__ W1,
    const float* __restrict__ b1, const float* __restrict__ W2,
    const float* __restrict__ b2, float* __restrict__ out)
{
    __shared__ float lds[3 * ROWPAD];

    const int tid   = threadIdx.x;
    const int bc    = blockIdx.x;          // b*64 + c
    const int b     = bc >> 6;
    const int c     = bc & 63;
    const int lane  = tid & 31;
    const int wave  = tid >> 5;
    const int ln    = lane & 15;           // N / M / sample index within half
    const int khalf = lane >> 4;           // which K/M half

    // ---- stage rows {c-1, c, c+1} (clamped) of X into LDS ----
    const int rm1 = (c > 0)  ? c - 1 : 0;
    const int rp1 = (c < 63) ? c + 1 : 63;
    const float4* x0 = (const float4*)(X + ((size_t)b * C_ + rm1) * L_);
    const float4* x1 = (const float4*)(X + ((size_t)b * C_ + c  ) * L_);
    const float4* x2 = (const float4*)(X + ((size_t)b * C_ + rp1) * L_);
    float4* l4 = (float4*)lds;
#if ASYNC_STAGE
    #pragma unroll
    for (int i = 0; i < (L_ / 4) / 256; ++i) {
        int idx = tid + i * 256;
        __builtin_amdgcn_global_load_async_to_lds_b128(
            (AS_GLOBAL v4i_*)(x0 + idx),
            (AS_LDS    v4i_*)(l4 + 0 * (ROWPAD / 4) + idx), 0, 0);
        __builtin_amdgcn_global_load_async_to_lds_b128(
            (AS_GLOBAL v4i_*)(x1 + idx),
            (AS_LDS    v4i_*)(l4 + 1 * (ROWPAD / 4) + idx), 0, 0);
        __builtin_amdgcn_global_load_async_to_lds_b128(
            (AS_GLOBAL v4i_*)(x2 + idx),
            (AS_LDS    v4i_*)(l4 + 2 * (ROWPAD / 4) + idx), 0, 0);
    }
#else
    #pragma unroll
    for (int i = 0; i < (L_ / 4) / 256; ++i) {
        int idx = tid + i * 256;
        l4[0 * (ROWPAD / 4) + idx] = x0[idx];
        l4[1 * (ROWPAD / 4) + idx] = x1[idx];
        l4[2 * (ROWPAD / 4) + idx] = x2[idx];
    }
#endif
    if (tid < 16) lds[ROWPAD + L_ + tid] = 0.0f;   // pad for patch 511

    // ---- preload W1 as WMMA B operands: bmat[kk][nt], lane holds (K-pair, N) ----
    v2f bmat[4][4];
    #pragma unroll
    for (int nt = 0; nt < 4; ++nt)
        #pragma unroll
        for (int kk = 0; kk < 4; ++kk)
            bmat[kk][nt] = *(const v2f*)(W1 + (nt * 16 + ln) * PS_ + kk * 4 + khalf * 2);

    float b1v[4], w20[4], w21[4];
    #pragma unroll
    for (int nt = 0; nt < 4; ++nt) {
        b1v[nt] = b1[nt * 16 + ln];
        w20[nt] = W2[0 * HID_ + nt * 16 + ln];
        w21[nt] = W2[1 * HID_ + nt * 16 + ln];
    }
    const float b20 = b2[0], b21 = b2[1];

#if ASYNC_STAGE
    __builtin_amdgcn_s_wait_asynccnt(0);   // my async LDS writes are visible
#endif
    __syncthreads();

    // ---- degenerate y interpolation (uniform per block, computed faithfully) ----
    float iyv  = ((float)c * (1.0f / 63.0f)) * 63.0f;
    float iy0f = floorf(iyv);
    float wy   = iyv - iy0f;
    int   iy0  = min(max((int)iy0f, 0), 63);
    int   iy1  = min(iy0 + 1, 63);
    int   reg0 = min(max(iy0 - c + 1, 0), 2);   // LDS row region for iy0
    int   reg1 = min(max(iy1 - c + 1, 0), 2);   // LDS row region for iy1

    float* row1 = lds + ROWPAD;                 // row c (patch source)
    const float inv4095 = 1.0f / 4095.0f;
    float* outbc = out + (size_t)bc * PC_ * PS_;

    const float t = (float)ln * (1.0f / 15.0f); // this lane's sample position

    #pragma unroll 1
    for (int tile = 0; tile < 4; ++tile) {
        const int p0 = (wave * 4 + tile) * 16;  // 16-patch tile base (0..496)

        // ---- layer 1: D = patches(16x16) x W1^T(16x64) via 16 WMMA f32 16x16x4 ----
        v8f acc[4];
        #pragma unroll
        for (int nt = 0; nt < 4; ++nt)
            #pragma unroll
            for (int e = 0; e < 8; ++e) acc[nt][e] = b1v[nt];

        #pragma unroll
        for (int kk = 0; kk < 4; ++kk) {
            v2f a = *(const v2f*)(row1 + (p0 + ln) * STR_ + kk * 4 + khalf * 2);
            #pragma unroll
            for (int nt = 0; nt < 4; ++nt)
                acc[nt] = __builtin_amdgcn_wmma_f32_16x16x4_f32(
                    false, a, false, bmat[kk][nt], (short)0, acc[nt], false, false);
        }

        // ---- exact gelu, branch-free ----
        #pragma unroll
        for (int nt = 0; nt < 4; ++nt)
            #pragma unroll
            for (int e = 0; e < 8; ++e)
                acc[nt][e] = gelu_exact(acc[nt][e]);

        // ---- layer 2 (64 -> 2): per-lane partials + 16-lane xor-shuffle reduce ----
        float red0[8], red1[8];
        #pragma unroll
        for (int r = 0; r < 8; ++r) {
            float s0 = 0.0f, s1 = 0.0f;
            #pragma unroll
            for (int nt = 0; nt < 4; ++nt) {
                float hv = acc[nt][r];
                s0 = fmaf(hv, w20[nt], s0);
                s1 = fmaf(hv, w21[nt], s1);
            }
            #pragma unroll
            for (int m = 1; m <= 8; m <<= 1) {   // stays within each 16-lane half
                s0 += __shfl_xor(s0, m, 32);
                s1 += __shfl_xor(s1, m, 32);
            }
            red0[r] = s0 + b20;                  // dx      for patch M = khalf*8 + r
            red1[r] = s1 + b21;                  // ds-pre  for patch M = khalf*8 + r
        }

        // ---- sampling: lane handles sample s=ln for the 8 patches of its half ----
        #pragma unroll
        for (int r = 0; r < 8; ++r) {
            int   p      = p0 + khalf * 8 + r;
            float anchor = (float)p * 8.0f + 7.5f;
            float dsv    = fmaxf(red1[r] + 7.5f, 0.0f);
            float ctr    = red0[r] + anchor;
            float lo     = fminf(fmaxf((ctr - dsv) * inv4095, 0.0f), 1.0f);
            float hi     = fminf(fmaxf((ctr + dsv) * inv4095, 0.0f), 1.0f);
            float xs     = lo + (hi - lo) * t;
            float ix     = xs * 4095.0f;
            float ix0f   = floorf(ix);
            float wx     = ix - ix0f;
            int   ix0    = min(max((int)ix0f, 0), 4095);
            int   ix1    = min(ix0 + 1, 4095);
            float v00 = lds[reg0 * ROWPAD + ix0];
            float v01 = lds[reg0 * ROWPAD + ix1];
            float v10 = lds[reg1 * ROWPAD + ix0];
            float v11 = lds[reg1 * ROWPAD + ix1];
            float top = v00 + (v01 - v00) * wx;
            float bot = v10 + (v11 - v10) * wx;
            float val = top + (bot - top) * wy;
            if (p < PC_) outbc[(size_t)p * PS_ + ln] = val;  // coalesced over s
        }
    }
}

extern "C" void kernel_launch(void* const* d_in, const int* in_sizes, int n_in,
                              void* d_out, int out_size, void* d_ws, size_t ws_size,
                              hipStream_t stream) {
    const float* X  = (const float*)d_in[0];
    const float* W1 = (const float*)d_in[1];
    const float* b1 = (const float*)d_in[2];
    const float* W2 = (const float*)d_in[3];
    const float* b2 = (const float*)d_in[4];
    float* out = (float*)d_out;
    depatch_sampling_kernel<<<dim3(B_ * C_), 256, 0, stream>>>(X, W1, b1, W2, b2, out);
}